// Attention_KVCache_41111426957847
// MI455X (gfx1250) — compile-verified
//
#include <hip/hip_runtime.h>

// Problem dims (fixed by reference)
#define B_   32
#define S_   4
#define D_   1024
#define H_   16
#define KV_  4096
#define HD_  64
#define L_   (KV_ + S_)      // 4100
#define EPS_ 1e-5f

typedef __attribute__((ext_vector_type(2))) float v2f;
typedef __attribute__((ext_vector_type(8))) float v8f;

// ---------------------------------------------------------------------------
// 1) QKV projection: C[128,1024] = x[128,1024] @ w^T + b, for wq/wk/wv.
//    One wave32 per 16x16 output tile, v_wmma_f32_16x16x4_f32 over K=1024.
//    A layout (16x4 f32): lanes 0-15 M=0..15, VGPR j holds K = j + 2*(lane/16)
//    B layout (4x16 f32): lanes 0-15 N=0..15, VGPR j holds K = j + 2*(lane/16)
//    C layout (16x16):    VGPR j -> M = j + 8*(lane/16), N = lane%16
// ---------------------------------------------------------------------------
__global__ __launch_bounds__(32) void qkv_proj_wmma(
    const float* __restrict__ x,
    const float* __restrict__ wq, const float* __restrict__ bq,
    const float* __restrict__ wk, const float* __restrict__ bk,
    const float* __restrict__ wv, const float* __restrict__ bv,
    float* __restrict__ q_ws, float* __restrict__ ko, float* __restrict__ vo)
{
  const int tilesN = D_ / 16;                 // 64
  const int tilesM = (B_ * S_) / 16;          // 8
  int t     = blockIdx.x;
  int which = t / (tilesM * tilesN);          // 0=q 1=k 2=v
  int rem   = t % (tilesM * tilesN);
  int m0    = (rem / tilesN) * 16;
  int n0    = (rem % tilesN) * 16;
  const float* w    = (which == 0) ? wq : (which == 1) ? wk : wv;
  const float* bias = (which == 0) ? bq : (which == 1) ? bk : bv;

  int lane = threadIdx.x;
  int g = lane >> 4, r = lane & 15;
  int am = m0 + r;                            // A row (M)
  int bn = n0 + r;                            // B col (N) -> w row
  v8f c = {};
  for (int k0 = 0; k0 < D_; k0 += 4) {
    int kb = k0 + 2 * g;
    v2f a = *(const v2f*)(x + (long long)am * D_ + kb);     // x[m, kb..kb+1]
    v2f b = *(const v2f*)(w + (long long)bn * D_ + kb);     // w[n, kb..kb+1]
    c = __builtin_amdgcn_wmma_f32_16x16x4_f32(false, a, false, b,
                                              (short)0, c, false, false);
  }

  int n  = n0 + r;
  int h  = n / HD_, d = n % HD_;
  float bb = bias[n];
  for (int j = 0; j < 8; ++j) {
    int m   = j + 8 * g;
    int row = m0 + m;                         // row = b*S + s
    int bb_ = row / S_, s = row % S_;
    float val = c[j] + bb;
    if (which == 0) {
      q_ws[(((long long)bb_ * H_ + h) * S_ + s) * HD_ + d] = val;       // [B,H,S,HD]
    } else {
      float* dst = (which == 1) ? ko : vo;    // append rows KV..KV+S-1
      dst[(((long long)bb_ * H_ + h) * L_ + (KV_ + s)) * HD_ + d] = val;
    }
  }
}

// ---------------------------------------------------------------------------
// 2) Fused flash attention + K/V-cache concat writer.
//    One 256-thread WG per (b,h). Streams 128-key chunks: stage K/V in LDS
//    (write-through to the concatenated k/v outputs so cache bytes are
//    touched exactly once), WMMA scores, online softmax, WMMA P@V with
//    persistent rescaled accumulators. Next chunk prefetched during compute.
// ---------------------------------------------------------------------------
#define CHUNK  128
#define NCHUNK ((L_ + CHUNK - 1) / CHUNK)     // 33
#define KPAD   (HD_ + 4)                      // 68: 16B-aligned rows, no bank conflicts

__global__ __launch_bounds__(256) void attn_fused(
    const float* __restrict__ cache_k, const float* __restrict__ cache_v,
    const float* __restrict__ q_ws,
    float* __restrict__ ko, float* __restrict__ vo,
    float* __restrict__ att_ws)
{
  __shared__ float Ks[CHUNK][KPAD];           // ~34 KB
  __shared__ float Vs[CHUNK][KPAD];           // ~34 KB
  __shared__ float sc[S_][CHUNK];             // 2 KB (scores -> p values)
  __shared__ float qs[S_][KPAD];              // ~1 KB
  __shared__ float row_max[S_], row_sum[S_], row_scale[S_];

  int bh  = blockIdx.x;                       // b*H + h
  int b_  = bh / H_, h = bh % H_;
  int tid = threadIdx.x;
  int wave = tid >> 5, lane = tid & 31;
  int g = lane >> 4, r = lane & 15;

  if (tid < S_ * HD_) qs[tid >> 6][tid & 63] = q_ws[(long long)bh * S_ * HD_ + tid];
  if (tid < S_) { row_max[tid] = -__builtin_inff(); row_sum[tid] = 0.0f; }

  const float* ckb = cache_k + (long long)bh * KV_ * HD_;
  const float* cvb = cache_v + (long long)bh * KV_ * HD_;
  float* kob = ko + (long long)bh * L_ * HD_;
  float* vob = vo + (long long)bh * L_ * HD_;

  v8f oacc = {};                              // P@V accumulator (waves 0..3)

  __syncthreads();

  for (int ch = 0; ch < NCHUNK; ++ch) {
    int base = ch * CHUNK;

    // ---- stage K/V chunk into LDS; write-through cache rows to outputs ----
    for (int i = tid; i < CHUNK * (HD_ / 4); i += 256) {
      int row  = i / (HD_ / 4);
      int col4 = i % (HD_ / 4);
      int n    = base + row;
      float4 kv4 = {0.f, 0.f, 0.f, 0.f}, vv4 = {0.f, 0.f, 0.f, 0.f};
      if (n < KV_) {
        kv4 = ((const float4*)ckb)[(long long)n * (HD_ / 4) + col4];
        vv4 = ((const float4*)cvb)[(long long)n * (HD_ / 4) + col4];
        ((float4*)kob)[(long long)n * (HD_ / 4) + col4] = kv4;   // concat write
        ((float4*)vob)[(long long)n * (HD_ / 4) + col4] = vv4;
      } else if (n < L_) {
        kv4 = ((const float4*)kob)[(long long)n * (HD_ / 4) + col4];  // new rows
        vv4 = ((const float4*)vob)[(long long)n * (HD_ / 4) + col4];
      }
      *(float4*)&Ks[row][col4 * 4] = kv4;
      *(float4*)&Vs[row][col4 * 4] = vv4;
    }

    // ---- prefetch next chunk (global_prefetch_b8) while we compute -------
    {
      int nbase = base + CHUNK;
      int prow  = tid >> 1;                   // 0..127
      int phalf = tid & 1;                    // two 128B halves of a 256B row
      int pn    = nbase + prow;
      if (pn < KV_) {
        const float* pk = ckb + (long long)pn * HD_ + phalf * 32;
        const float* pv = cvb + (long long)pn * HD_ + phalf * 32;
        __builtin_prefetch(pk, 0, 0);
        __builtin_prefetch(pv, 0, 0);
      }
    }
    __syncthreads();

    // ---- scores: wave w computes keys [16w, 16w+16) of the chunk ---------
    {
      int krow = wave * 16 + r;               // local key index
      v8f c = {};
      for (int k0 = 0; k0 < HD_; k0 += 4) {
        int kk = k0 + 2 * g;
        v2f a  = *(const v2f*)&qs[r & 3][kk];        // rows 4..15 dup of 0..3
        v2f bb = *(const v2f*)&Ks[krow][kk];
        c = __builtin_amdgcn_wmma_f32_16x16x4_f32(false, a, false, bb,
                                                  (short)0, c, false, false);
      }
      if (g == 0) {
        int n = base + krow;
        for (int j = 0; j < S_; ++j)
          sc[j][krow] = (n < L_) ? c[j] * 0.125f      // 1/sqrt(HD)
                                 : -__builtin_inff();
      }
    }
    __syncthreads();

    // ---- online softmax: wave m owns query row m -------------------------
    if (wave < S_) {
      int m = wave;
      float lmax = -__builtin_inff();
      for (int i = lane; i < CHUNK; i += 32) lmax = fmaxf(lmax, sc[m][i]);
      for (int off = 16; off > 0; off >>= 1)
        lmax = fmaxf(lmax, __shfl_xor(lmax, off, 32));
      float newmax = fmaxf(row_max[m], lmax);
      float lsum = 0.0f;
      for (int i = lane; i < CHUNK; i += 32) {
        float e = __expf(sc[m][i] - newmax);
        sc[m][i] = e;                           // store p values
        lsum += e;
      }
      for (int off = 16; off > 0; off >>= 1)
        lsum += __shfl_xor(lsum, off, 32);
      if (lane == 0) {
        float f = __expf(row_max[m] - newmax);
        row_scale[m] = f;
        row_sum[m]   = row_sum[m] * f + lsum;
        row_max[m]   = newmax;
      }
    }
    __syncthreads();

    // ---- P @ V via WMMA: wave nt (0..3) owns output cols [16nt,16nt+16) --
    // O = P(4x128, padded M=16) x V(128x64). Accumulator persists across
    // chunks; rescaled by the online-softmax factor each chunk.
    if (wave < 4) {
      float f = row_scale[0];                 // rescale per C row m = j (mod 4)
      float f1 = row_scale[1], f2 = row_scale[2], f3 = row_scale[3];
      oacc[0] *= f;  oacc[4] *= f;
      oacc[1] *= f1; oacc[5] *= f1;
      oacc[2] *= f2; oacc[6] *= f2;
      oacc[3] *= f3; oacc[7] *= f3;
      int n0 = wave * 16;
      for (int kb = 0; kb < CHUNK; kb += 4) {
        int kk = kb + 2 * g;
        v2f a = *(const v2f*)&sc[r & 3][kk];  // P rows (dup for m>=4)
        v2f b;
        b.x = Vs[kk][n0 + r];                 // B[k][n] = V[k][n0+n]
        b.y = Vs[kk + 1][n0 + r];
        oacc = __builtin_amdgcn_wmma_f32_16x16x4_f32(false, a, false, b,
                                                     (short)0, oacc, false, false);
      }
    }
    __syncthreads();                          // sc/Ks/Vs reused next chunk
  }

  // ---- writeout: waves 0..3, C rows m=0..3 live in VGPRs 0..3 of g==0 ----
  if (wave < 4 && g == 0) {
    int n0 = wave * 16;
    for (int j = 0; j < S_; ++j) {
      float val = oacc[j] / row_sum[j];
      att_ws[((long long)b_ * S_ + j) * D_ + h * HD_ + n0 + r] = val;
    }
  }
}

// ---------------------------------------------------------------------------
// 3) LayerNorm over D=1024 per (b,s) row.
// ---------------------------------------------------------------------------
__global__ __launch_bounds__(256) void layernorm_kernel(
    const float* __restrict__ att_ws,
    const float* __restrict__ gamma, const float* __restrict__ beta,
    float* __restrict__ out)
{
  __shared__ float red[256];
  int row = blockIdx.x;
  int tid = threadIdx.x;
  const float* xr = att_ws + (long long)row * D_;

  float s = 0.f, s2 = 0.f, vals[4];
  for (int i = 0; i < 4; ++i) {
    float v = xr[tid + i * 256];
    vals[i] = v; s += v; s2 += v * v;
  }
  red[tid] = s; __syncthreads();
  for (int st = 128; st > 0; st >>= 1) { if (tid < st) red[tid] += red[tid + st]; __syncthreads(); }
  float mean = red[0] * (1.0f / D_); __syncthreads();
  red[tid] = s2; __syncthreads();
  for (int st = 128; st > 0; st >>= 1) { if (tid < st) red[tid] += red[tid + st]; __syncthreads(); }
  float var = red[0] * (1.0f / D_) - mean * mean;
  float inv = rsqrtf(var + EPS_);
  for (int i = 0; i < 4; ++i) {
    int idx = tid + i * 256;
    out[(long long)row * D_ + idx] = (vals[i] - mean) * inv * gamma[idx] + beta[idx];
  }
}

// ---------------------------------------------------------------------------
extern "C" void kernel_launch(void* const* d_in, const int* in_sizes, int n_in,
                              void* d_out, int out_size, void* d_ws, size_t ws_size,
                              hipStream_t stream) {
  (void)in_sizes; (void)n_in; (void)out_size; (void)ws_size;

  const float* x     = (const float*)d_in[0];
  const float* ck    = (const float*)d_in[1];
  const float* cv    = (const float*)d_in[2];
  const float* wq    = (const float*)d_in[3];
  const float* bq    = (const float*)d_in[4];
  const float* wk    = (const float*)d_in[5];
  const float* bk    = (const float*)d_in[6];
  const float* wv    = (const float*)d_in[7];
  const float* bv    = (const float*)d_in[8];
  const float* gamma = (const float*)d_in[9];
  const float* beta  = (const float*)d_in[10];

  float* out = (float*)d_out;                               // [B,S,D]
  float* ko  = out + (long long)B_ * S_ * D_;               // [B,H,L,HD]
  float* vo  = ko  + (long long)B_ * H_ * L_ * HD_;         // [B,H,L,HD]

  float* q_ws   = (float*)d_ws;                             // [B,H,S,HD]
  float* att_ws = q_ws + (long long)B_ * H_ * S_ * HD_;     // [B,S,D]

  qkv_proj_wmma<<<dim3(3 * 8 * (D_ / 16)), dim3(32), 0, stream>>>(
      x, wq, bq, wk, bk, wv, bv, q_ws, ko, vo);
  attn_fused<<<dim3(B_ * H_), dim3(256), 0, stream>>>(ck, cv, q_ws, ko, vo, att_ws);
  layernorm_kernel<<<dim3(B_ * S_), dim3(256), 0, stream>>>(att_ws, gamma, beta, out);
}